// BiLSTM_CRF_59047210385780
// MI455X (gfx1250) — compile-verified
//
#include <hip/hip_runtime.h>
#include <hip/hip_bf16.h>
#include <math.h>

constexpr int SEQ  = 8192;
constexpr int EMB  = 256;
constexpr int HID  = 512;
constexpr int Hd   = 256;    // per-direction hidden
constexpr int G4   = 1024;   // 4*Hd
constexpr int NTAG = 20;
constexpr int KCH  = Hd / 4; // 64 k-chunks of 4 in the recurrence matvec
constexpr int LCH  = 16;     // k-chunks of Whh_c cached in LDS (256 KB)

typedef __attribute__((ext_vector_type(2))) float v2f;
typedef __attribute__((ext_vector_type(8))) float v8f;

// ---------------- 1. embedding gather: x[s] = embed[sentence[s]] ----------------
__global__ void gather_embed(const int* __restrict__ sent,
                             const float* __restrict__ embed,
                             float* __restrict__ x) {
    const int s   = blockIdx.x;
    const int tok = sent[s];
    const float4* src = (const float4*)(embed + (long)tok * EMB);
    float4*       dst = (float4*)(x + (long)s * EMB);
    dst[threadIdx.x] = src[threadIdx.x];   // 64 threads x float4 = 256 floats
}

// ---------------- 2. input projection GEMM via f32 WMMA ----------------
// xg[s, j] = sum_e x[s,e] * Wih[j,e] + (bih[j] + bhh[j]),  both directions.
// One wave (32 lanes) computes a 16x16 tile; K=256 consumed in 64 chunks of 4.
__global__ void __launch_bounds__(32)
inproj_wmma(const float* __restrict__ x,
            const float* __restrict__ Wih_f,
            const float* __restrict__ bih_f,
            const float* __restrict__ bhh_f,
            const float* __restrict__ Wih_b,
            const float* __restrict__ bih_b,
            const float* __restrict__ bhh_b,
            float* __restrict__ xg_f,
            float* __restrict__ xg_b) {
    const int dir = blockIdx.z;
    const float* W  = dir ? Wih_b : Wih_f;
    const float* b0 = dir ? bih_b : bih_f;
    const float* b1 = dir ? bhh_b : bhh_f;
    float*       out = dir ? xg_b : xg_f;

    const int s0 = blockIdx.x * 16;
    const int j0 = blockIdx.y * 16;
    const int lane  = threadIdx.x;
    const int m     = lane & 15;     // row (A) / column (B,D) index
    const int khalf = lane >> 4;     // 0: K={0,1}, 1: K={2,3} within a chunk

    // ISA 16x4 f32 A layout: lanes 0-15 hold K=0(v0),K=1(v1); lanes 16-31 K=2,K=3.
    // B 4x16 layout is symmetric (row K striped across lanes within a VGPR).
    const float* xrow = x + (long)(s0 + m) * EMB + 2 * khalf;
    const float* wrow = W + (long)(j0 + m) * EMB + 2 * khalf;

    v8f acc = {};
    #pragma unroll 4
    for (int kc = 0; kc < EMB / 4; ++kc) {
        v2f a, bm;
        a.x  = xrow[kc * 4 + 0];
        a.y  = xrow[kc * 4 + 1];
        bm.x = wrow[kc * 4 + 0];
        bm.y = wrow[kc * 4 + 1];
        acc = __builtin_amdgcn_wmma_f32_16x16x4_f32(
                  false, a, false, bm, (short)0, acc, false, false);
    }

    const float bias = b0[j0 + m] + b1[j0 + m];
    #pragma unroll
    for (int i = 0; i < 8; ++i) {
        const int row = i + 8 * khalf;                 // C/D: vgpr i -> M=i (+8 for hi lanes)
        out[(long)(s0 + row) * G4 + (j0 + m)] = acc[i] + bias;
    }
}

// ---------------- 3a. re-layout Whh for coalesced recurrence loads ----------------
// Whh_c[((k4 * G4) + j) * 4 + kk] = Whh[j * Hd + k4*4 + kk]
// -> in the recurrence, lane j loads float4 at (k4*G4 + j)*16 B: 32 lanes = one
//    contiguous 512 B transaction instead of 32 scattered cachelines.
__global__ void whh_reorder(const float* __restrict__ Whh_f,
                            const float* __restrict__ Whh_b,
                            float* __restrict__ out_f,
                            float* __restrict__ out_b) {
    const int dir = blockIdx.z;
    const float* W   = dir ? Whh_b : Whh_f;
    float*       out = dir ? out_b : out_f;
    const int k4 = blockIdx.x;                                   // 0..KCH-1
    const int j  = blockIdx.y * blockDim.x + threadIdx.x;        // 0..G4-1
    const float4 v = *(const float4*)(W + (long)j * Hd + k4 * 4);
    *(float4*)(out + (((long)k4 * G4) + j) * 4) = v;
}

// ---------------- 3b. LSTM recurrence (sequential; one WG per direction) ----------------
__device__ __forceinline__ float sigmoidf_(float v) { return 1.0f / (1.0f + expf(-v)); }

// Dynamic LDS layout: [ whh cache: LCH*G4*4 floats (256 KB) | h: Hd | c: Hd | gates: G4 ]
constexpr size_t REC_LDS_FLOATS = (size_t)LCH * G4 * 4 + Hd + Hd + G4;
constexpr size_t REC_LDS_BYTES  = REC_LDS_FLOATS * sizeof(float);   // 268,288 B < 320 KB

__global__ void __launch_bounds__(1024)
lstm_recurrence(const float* __restrict__ xg_f,
                const float* __restrict__ xg_b,
                const float* __restrict__ whhc_f,
                const float* __restrict__ whhc_b,
                float* __restrict__ lstm_out) {
    extern __shared__ float smem[];
    float* whh_lds = smem;                         // LCH*G4*4 floats
    float* h       = smem + (size_t)LCH * G4 * 4;  // Hd floats
    float* c       = h + Hd;                       // Hd floats
    float* gates   = c + Hd;                       // G4 floats

    const int dir = blockIdx.x;
    const float* xg   = dir ? xg_b   : xg_f;
    const float* whhc = dir ? whhc_b : whhc_f;

    const int j = threadIdx.x;                     // gate row 0..1023

    // one-time cooperative preload of LDS-resident weight chunks (coalesced)
    {
        const float4* src = (const float4*)whhc;
        float4*       dst = (float4*)whh_lds;
        for (int i = j; i < LCH * G4; i += 1024) dst[i] = src[i];
    }
    if (j < Hd) { h[j] = 0.0f; c[j] = 0.0f; }
    __syncthreads();

    const float4* wl = (const float4*)whh_lds + j;                 // + k4*G4 per chunk
    const float4* wg = (const float4*)whhc + (long)LCH * G4 + j;   // streamed part
    const float4* hv = (const float4*)h;

    for (int step = 0; step < SEQ; ++step) {
        const int t = dir ? (SEQ - 1 - step) : step;

        float acc = xg[(long)t * G4 + j];          // precomputed x-proj + bias
        #pragma unroll 4
        for (int k4 = 0; k4 < LCH; ++k4) {         // LDS-resident weights
            const float4 w  = wl[(size_t)k4 * G4];
            const float4 hh = hv[k4];
            acc += w.x * hh.x + w.y * hh.y + w.z * hh.z + w.w * hh.w;
        }
        #pragma unroll 4
        for (int k4 = 0; k4 < KCH - LCH; ++k4) {   // L2-streamed weights, coalesced
            const float4 w  = wg[(size_t)k4 * G4];
            const float4 hh = hv[LCH + k4];
            acc += w.x * hh.x + w.y * hh.y + w.z * hh.z + w.w * hh.w;
        }
        gates[j] = acc;

        if (step + 1 < SEQ) {                      // global_prefetch_b8 of next xg row
            const int tn = dir ? (SEQ - 2 - step) : (step + 1);
            __builtin_prefetch(xg + (long)tn * G4 + j, 0, 0);
        }
        __syncthreads();

        if (j < Hd) {                              // PyTorch gate order i,f,g,o
            const float ig = sigmoidf_(gates[j]);
            const float fg = sigmoidf_(gates[Hd + j]);
            const float gg = tanhf(gates[2 * Hd + j]);
            const float og = sigmoidf_(gates[3 * Hd + j]);
            const float cn = fg * c[j] + ig * gg;
            const float hn = og * tanhf(cn);
            c[j] = cn;
            h[j] = hn;
            lstm_out[(long)t * HID + dir * Hd + j] = hn;
        }
        __syncthreads();
    }
}

// ---------------- 4. emissions = lstm_out @ Wout^T + bout ----------------
__global__ void emissions_kernel(const float* __restrict__ lstm_out,
                                 const float* __restrict__ Wout,
                                 const float* __restrict__ bout,
                                 float* __restrict__ em) {
    const int idx = blockIdx.x * blockDim.x + threadIdx.x;
    if (idx >= SEQ * NTAG) return;
    const int s = idx / NTAG, t = idx % NTAG;
    const float4* a = (const float4*)(lstm_out + (long)s * HID);
    const float4* w = (const float4*)(Wout + (long)t * HID);
    float acc = bout[t];
    #pragma unroll 8
    for (int k = 0; k < HID / 4; ++k) {
        const float4 av = a[k], wv = w[k];
        acc += av.x * wv.x + av.y * wv.y + av.z * wv.z + av.w * wv.w;
    }
    em[idx] = acc;
}

// ---------------- 5. Viterbi decode (one wave32) ----------------
__global__ void __launch_bounds__(32)
viterbi_kernel(const float* __restrict__ em,
               const float* __restrict__ start_trans,
               const float* __restrict__ end_trans,
               const float* __restrict__ trans,
               int* __restrict__ bp,
               int* __restrict__ path) {
    __shared__ float score[NTAG];
    __shared__ float tr[NTAG * NTAG];
    const int tid = threadIdx.x;
    for (int i = tid; i < NTAG * NTAG; i += 32) tr[i] = trans[i];
    if (tid < NTAG) score[tid] = start_trans[tid] + em[tid];
    __syncthreads();

    for (int s = 1; s < SEQ; ++s) {
        float best = -3.4e38f; int argp = 0;
        if (tid < NTAG) {
            #pragma unroll
            for (int p = 0; p < NTAG; ++p) {
                const float v = score[p] + tr[p * NTAG + tid];
                if (v > best) { best = v; argp = p; }   // strict > = argmax first-max
            }
            best += em[s * NTAG + tid];
            bp[(s - 1) * NTAG + tid] = argp;
        }
        __syncthreads();
        if (tid < NTAG) score[tid] = best;
        __syncthreads();
    }

    if (tid == 0) {
        float bestv = -3.4e38f; int last = 0;
        for (int t2 = 0; t2 < NTAG; ++t2) {
            const float v = score[t2] + end_trans[t2];
            if (v > bestv) { bestv = v; last = t2; }
        }
        path[SEQ - 1] = last;
        for (int s = SEQ - 2; s >= 0; --s) {
            last = bp[s * NTAG + last];
            path[s] = last;
        }
    }
}

// ---------------- host-side launch ----------------
extern "C" void kernel_launch(void* const* d_in, const int* in_sizes, int n_in,
                              void* d_out, int out_size, void* d_ws, size_t ws_size,
                              hipStream_t stream) {
    const int*   sentence = (const int*)  d_in[0];
    const float* embed    = (const float*)d_in[1];
    const float* Wih_f = (const float*)d_in[2];
    const float* Whh_f = (const float*)d_in[3];
    const float* bih_f = (const float*)d_in[4];
    const float* bhh_f = (const float*)d_in[5];
    const float* Wih_b = (const float*)d_in[6];
    const float* Whh_b = (const float*)d_in[7];
    const float* bih_b = (const float*)d_in[8];
    const float* bhh_b = (const float*)d_in[9];
    const float* Wout  = (const float*)d_in[10];
    const float* bout  = (const float*)d_in[11];
    const float* start_trans = (const float*)d_in[12];
    const float* end_trans   = (const float*)d_in[13];
    const float* trans       = (const float*)d_in[14];

    char* ws = (char*)d_ws;
    size_t off = 0;
    auto carve = [&](size_t bytes) -> char* {
        char* p = ws + off;
        off += (bytes + 255) & ~(size_t)255;
        return p;
    };
    float* x       = (float*)carve(sizeof(float) * SEQ * EMB);
    float* xg_f    = (float*)carve(sizeof(float) * SEQ * G4);
    float* xg_b    = (float*)carve(sizeof(float) * SEQ * G4);
    float* whhc_f  = (float*)carve(sizeof(float) * G4 * Hd);
    float* whhc_b  = (float*)carve(sizeof(float) * G4 * Hd);
    float* lstmout = (float*)carve(sizeof(float) * SEQ * HID);
    float* em      = (float*)carve(sizeof(float) * SEQ * NTAG);
    int*   bp      = (int*)  carve(sizeof(int)   * SEQ * NTAG);
    int*   path    = (int*)d_out;   // reference output dtype is int32

    gather_embed<<<SEQ, 64, 0, stream>>>(sentence, embed, x);

    dim3 gproj(SEQ / 16, G4 / 16, 2);
    inproj_wmma<<<gproj, 32, 0, stream>>>(x, Wih_f, bih_f, bhh_f,
                                          Wih_b, bih_b, bhh_b, xg_f, xg_b);

    dim3 gre(KCH, G4 / 256, 2);
    whh_reorder<<<gre, 256, 0, stream>>>(Whh_f, Whh_b, whhc_f, whhc_b);

    lstm_recurrence<<<2, 1024, REC_LDS_BYTES, stream>>>(xg_f, xg_b,
                                                        whhc_f, whhc_b, lstmout);

    emissions_kernel<<<(SEQ * NTAG + 255) / 256, 256, 0, stream>>>(lstmout, Wout, bout, em);

    viterbi_kernel<<<1, 32, 0, stream>>>(em, start_trans, end_trans, trans, bp, path);
}